// InvariantPolynomial_49125835932145
// MI455X (gfx1250) — compile-verified
//
#include <hip/hip_runtime.h>
#include <math.h>

typedef float v2f __attribute__((ext_vector_type(2)));
typedef float v8f __attribute__((ext_vector_type(8)));

#define INV_SQRT2 0.70710678118654752f

// ---------------------------------------------------------------------------
// Spherical harmonics l=0..3 (exactly mirrors reference sph_harm3)
// ---------------------------------------------------------------------------
__device__ __forceinline__ void calc_sh(float x, float y, float z, float* sh) {
  const float s3 = 1.7320508075688772f;
  const float s5 = 2.23606797749979f;
  const float s7 = 2.6457513110645907f;
  const float c56 = 0.9128709291752769f;   // sqrt(5/6)
  const float c38 = 0.6123724356957945f;   // sqrt(3/8)
  float x2 = x * x, y2 = y * y, z2 = z * z;
  float x2z2 = x2 + z2;
  float sh2_0 = s3 * x * z;
  float sh2_1 = s3 * x * y;
  float sh2_2 = y2 - 0.5f * x2z2;
  float sh2_3 = s3 * y * z;
  float sh2_4 = 0.5f * s3 * (z2 - x2);
  sh[0] = 1.f;
  sh[1] = s3 * x; sh[2] = s3 * y; sh[3] = s3 * z;
  sh[4] = s5 * sh2_0; sh[5] = s5 * sh2_1; sh[6] = s5 * sh2_2;
  sh[7] = s5 * sh2_3; sh[8] = s5 * sh2_4;
  float sh3_0 = c56 * (sh2_0 * z + sh2_4 * x);
  float sh3_1 = s5 * sh2_0 * y;
  float sh3_2 = c38 * (4.f * y2 - x2z2) * x;
  float sh3_3 = 0.5f * y * (2.f * y2 - 3.f * x2z2);
  float sh3_4 = c38 * z * (4.f * y2 - x2z2);
  float sh3_5 = s5 * sh2_4 * y;
  float sh3_6 = c56 * (sh2_4 * z - sh2_0 * x);
  sh[9] = s7 * sh3_0; sh[10] = s7 * sh3_1; sh[11] = s7 * sh3_2;
  sh[12] = s7 * sh3_3; sh[13] = s7 * sh3_4; sh[14] = s7 * sh3_5;
  sh[15] = s7 * sh3_6;
}

// ---------------------------------------------------------------------------
// Runtime Wigner-3j builder (faithful port of reference _su2_cg/_q/_w3j).
// Table: slot (l1*4+l2)*3+lo, element (i*7+j)*5+k, 48 slots * 245 floats.
// ---------------------------------------------------------------------------
__device__ double dfact(int n) { double r = 1.0; for (int i = 2; i <= n; i++) r *= (double)i; return r; }

__device__ void make_q(int l, double qre[7][7], double qim[7][7]) {
  for (int i = 0; i < 7; i++) for (int j = 0; j < 7; j++) { qre[i][j] = 0.0; qim[i][j] = 0.0; }
  const double is2 = 0.7071067811865476;
  for (int m = -l; m < 0; m++) {
    qre[l + m][l - m] = is2;      // q[l+m][l+|m|] = 1/sqrt2
    qim[l + m][l + m] = -is2;     // q[l+m][l-|m|] = -i/sqrt2
  }
  qre[l][l] = 1.0;
  for (int m = 1; m <= l; m++) {
    double sg = (m & 1) ? -1.0 : 1.0;
    qre[l + m][l + m] = sg * is2;
    qim[l + m][l - m] = sg * is2; // i*(-1)^m/sqrt2
  }
  double fre, fim;                // (-i)^l
  switch (l & 3) { case 0: fre = 1; fim = 0; break; case 1: fre = 0; fim = -1; break;
                   case 2: fre = -1; fim = 0; break; default: fre = 0; fim = 1; }
  for (int i = 0; i < 7; i++) for (int j = 0; j < 7; j++) {
    double a = qre[i][j], b = qim[i][j];
    qre[i][j] = a * fre - b * fim;
    qim[i][j] = a * fim + b * fre;
  }
}

__global__ void k_build_w3j(float* w3j) {
  int t = blockIdx.x * blockDim.x + threadIdx.x;
  if (t >= 48) return;
  int l1 = t / 12, l2 = (t % 12) / 3, lo = t % 3;
  float* out = w3j + t * 245;
  for (int i = 0; i < 245; i++) out[i] = 0.f;
  if (lo < l1 - l2 || lo < l2 - l1 || lo > l1 + l2) return;

  int n1 = 2 * l1 + 1, n2 = 2 * l2 + 1, n3 = 2 * lo + 1;
  double C[7][7][5];
  for (int i = 0; i < 7; i++) for (int j = 0; j < 7; j++) for (int k = 0; k < 5; k++) C[i][j][k] = 0.0;
  for (int m1 = -l1; m1 <= l1; m1++) for (int m2 = -l2; m2 <= l2; m2++) {
    int m3 = m1 + m2;
    if (m3 < -lo || m3 > lo) continue;
    double pref = sqrt((2.0 * lo + 1) * dfact(l1 + l2 - lo) * dfact(l1 - l2 + lo) *
                       dfact(-l1 + l2 + lo) / dfact(l1 + l2 + lo + 1));
    pref *= sqrt(dfact(lo + m3) * dfact(lo - m3) * dfact(l1 - m1) * dfact(l1 + m1) *
                 dfact(l2 - m2) * dfact(l2 + m2));
    double s = 0.0;
    for (int k = 0; k <= l1 + l2 - lo; k++) {
      int d0 = k, d1 = l1 + l2 - lo - k, d2 = l1 - m1 - k, d3 = l2 + m2 - k;
      int d4 = lo - l2 + m1 + k, d5 = lo - l1 - m2 + k;
      if (d0 < 0 || d1 < 0 || d2 < 0 || d3 < 0 || d4 < 0 || d5 < 0) continue;
      double term = 1.0 / (dfact(d0) * dfact(d1) * dfact(d2) * dfact(d3) * dfact(d4) * dfact(d5));
      s += (k & 1) ? -term : term;
    }
    C[l1 + m1][l2 + m2][lo + m3] = pref * s;
  }
  double q1re[7][7], q1im[7][7], q2re[7][7], q2im[7][7], q3re[7][7], q3im[7][7];
  make_q(l1, q1re, q1im); make_q(l2, q2re, q2im); make_q(lo, q3re, q3im);
  double Tre[7][7][5], Tim[7][7][5];
  for (int i = 0; i < 7; i++) for (int j = 0; j < 7; j++) for (int k = 0; k < 5; k++) { Tre[i][j][k] = 0.0; Tim[i][j][k] = 0.0; }
  // T[j,l,m] = sum_{i,k,n} q1[i,j] q2[k,l] conj(q3[n,m]) C[i,k,n]
  for (int i = 0; i < n1; i++) for (int kk = 0; kk < n2; kk++) for (int n = 0; n < n3; n++) {
    double c = C[i][kk][n];
    if (c == 0.0) continue;
    for (int j = 0; j < n1; j++) {
      double are = q1re[i][j] * c, aim = q1im[i][j] * c;
      if (are == 0.0 && aim == 0.0) continue;
      for (int l = 0; l < n2; l++) {
        double bre = q2re[kk][l], bim = q2im[kk][l];
        double abre = are * bre - aim * bim, abim = are * bim + aim * bre;
        if (abre == 0.0 && abim == 0.0) continue;
        for (int m = 0; m < n3; m++) {
          double dre = q3re[n][m], dim = -q3im[n][m];  // conj
          Tre[j][l][m] += abre * dre - abim * dim;
          Tim[j][l][m] += abre * dim + abim * dre;
        }
      }
    }
  }
  double nr = 0.0, ni = 0.0;
  for (int j = 0; j < n1; j++) for (int l = 0; l < n2; l++) for (int m = 0; m < n3; m++) {
    nr += Tre[j][l][m] * Tre[j][l][m];
    ni += Tim[j][l][m] * Tim[j][l][m];
  }
  bool useRe = (nr >= ni);
  double inv = 1.0 / sqrt(useRe ? nr : ni);
  for (int j = 0; j < n1; j++) for (int l = 0; l < n2; l++) for (int m = 0; m < n3; m++)
    out[(j * 7 + l) * 5 + m] = (float)((useRe ? Tre[j][l][m] : Tim[j][l][m]) * inv);
}

// ---------------------------------------------------------------------------
// Weight preprocessing: fold alpha & layer scalings into WMMA B-matrices.
// w1m layout: per output irrep c, [Ppad][MOpad] row-major (K-major rows).
//   offsets {0,256,384,640,768}, total 832 floats.
// w2m layout: [144][16] (K=144 s-values -> N=16, cols 0..6 valid), 2304 floats.
// ---------------------------------------------------------------------------
__global__ void k_prep_weights(const float* __restrict__ w1, const float* __restrict__ w2,
                               float* __restrict__ w1m, float* __restrict__ w2m) {
  int t = blockIdx.x * blockDim.x + threadIdx.x;
  if (t < 832) {
    const int base[6]  = {0, 256, 384, 640, 768, 832};
    const int MOpad[5] = {64, 32, 32, 16, 16};
    const int Pcnt[5]  = {4, 3, 6, 7, 4};
    const int MO[5]    = {64, 24, 24, 16, 16};
    // alpha_c * (1/sqrt(2)) :  0.5, 1, sqrt(1/2), sqrt(5/7), sqrt(5/4)  (x 0.7071...)
    const float aS[5] = {0.35355339059f, 0.70710678119f, 0.5f, 0.59761430467f, 0.79056941504f};
    int c = 0; while (t >= base[c + 1]) c++;
    int rel = t - base[c];
    int prow = rel / MOpad[c], col = rel % MOpad[c];
    float v = 0.f;
    if (prow < Pcnt[c] && col < MO[c]) {
      const int woff0[4] = {0, 128, 344, 544};
      const int woff1[3] = {192, 408, 608};
      const int woff2[6] = {64, 104, 232, 304, 448, 504};
      const int woff3[7] = {88, 216, 272, 288, 432, 488, 632};
      const int woff4[4] = {256, 328, 472, 528};
      int wo;
      switch (c) { case 0: wo = woff0[prow]; break; case 1: wo = woff1[prow]; break;
                   case 2: wo = woff2[prow]; break; case 3: wo = woff3[prow]; break;
                   default: wo = woff4[prow]; }
      v = aS[c] * w1[wo + col];
    }
    w1m[t] = v;
  } else if (t < 832 + 2304) {
    int r = t - 832;
    int row = r / 16, col = r % 16;
    // sc2 = (1/sqrt2)*(1/2);  alpha0 = sqrt(1/40); alpha1 = sqrt(1/104)
    const float s0 = 0.35355339059f * 0.15811388301f;
    const float s1 = 0.35355339059f * 0.09805806757f;
    float v = 0.f;
    if (col == 0) {
      if (row < 24) v = s0 * w2[384 + row];                       // q2 (24,1,+)
      else if (row < 40) v = s0 * w2[648 + (row - 24)];           // q5 (16,2,-)
    } else if (col < 7) {
      int w = col - 1;
      if (row >= 40 && row < 104) v = s1 * w2[(row - 40) * 6 + w];        // q1 (64,0,+)
      else if (row >= 104 && row < 128) v = s1 * w2[408 + (row - 104) * 6 + w]; // q3 (24,1,-)
      else if (row >= 128 && row < 144) v = s1 * w2[552 + (row - 128) * 6 + w]; // q4 (16,2,+)
    }
    w2m[r] = v;
  }
}

// ---------------------------------------------------------------------------
// Stage 1: edge SH + scatter to nodeA (1/sqrt2 folded in)
// ---------------------------------------------------------------------------
__global__ void k_sh_scatter(const float* __restrict__ pos, const int* __restrict__ src,
                             const int* __restrict__ dst, float* __restrict__ nodeA, int E) {
  int e = blockIdx.x * blockDim.x + threadIdx.x;
  if (e >= E) return;
  int s = src[e], d = dst[e];
  float x = pos[3 * s] - pos[3 * d];
  float y = pos[3 * s + 1] - pos[3 * d + 1];
  float z = pos[3 * s + 2] - pos[3 * d + 2];
  float sh[16];
  calc_sh(x, y, z, sh);
  float* na = nodeA + (size_t)d * 16;
  for (int i = 0; i < 16; i++) atomicAdd(na + i, sh[i] * INV_SQRT2);
}

// ---------------------------------------------------------------------------
// Templated WMMA fan-out for one output irrep of TP1. All shape metadata is
// compile-time: B-fragments hoisted & reused across (k, half) tiles, two
// accumulator chains (M-halves) per B-fragment, bounds checks folded.
//   NK = 2*lo+1, NT = N-tiles, PPAD = padded path count (K), MO = multiplicity,
//   OO = output channel offset, AOFF = LDS A-stage offset, W1O = weight offset.
// ---------------------------------------------------------------------------
template <int NK, int NT, int PPAD, int MO, int OO, int AOFF, int W1O>
__device__ __forceinline__ void tp1_fanout(const float* __restrict__ A,
                                           const float* __restrict__ w1m,
                                           const int* __restrict__ D,
                                           float* __restrict__ nodeB,
                                           int h, int m) {
  constexpr int KS = PPAD / 4;
  constexpr int MOpad = NT * 16;
  const float* W = w1m + W1O;
#pragma unroll
  for (int nt = 0; nt < NT; nt++) {
    v2f b[KS];
#pragma unroll
    for (int ks = 0; ks < KS; ks++) {
      int K0 = ks * 4 + 2 * h;
      b[ks][0] = W[K0 * MOpad + nt * 16 + m];
      b[ks][1] = W[(K0 + 1) * MOpad + nt * 16 + m];
    }
    int col = nt * 16 + m;
    bool colok = (nt * 16 + 16 <= MO) ? true : (col < MO);  // folds when full tile
#pragma unroll
    for (int k = 0; k < NK; k++) {
      v8f acc0 = {}, acc1 = {};
#pragma unroll
      for (int ks = 0; ks < KS; ks++) {
        int K0 = ks * 4 + 2 * h;
        const float* Ap0 = A + AOFF + (k * 32 + m) * PPAD;
        const float* Ap1 = A + AOFF + (k * 32 + 16 + m) * PPAD;
        v2f a0, a1;
        a0[0] = Ap0[K0]; a0[1] = Ap0[K0 + 1];
        a1[0] = Ap1[K0]; a1[1] = Ap1[K0 + 1];
        acc0 = __builtin_amdgcn_wmma_f32_16x16x4_f32(false, a0, false, b[ks], (short)0, acc0,
                                                     false, false);
        acc1 = __builtin_amdgcn_wmma_f32_16x16x4_f32(false, a1, false, b[ks], (short)0, acc1,
                                                     false, false);
      }
#pragma unroll
      for (int i = 0; i < 8; i++) {
        int r = i + 8 * h;                 // C/D layout: vgpr i, halfwave h -> row
        int d0 = D[r];
        int d1 = D[16 + r];
        if (colok && d0 >= 0)
          atomicAdd(nodeB + (size_t)d0 * 368 + OO + col * NK + k, acc0[i]);
        if (colok && d1 >= 0)
          atomicAdd(nodeB + (size_t)d1 * 368 + OO + col * NK + k, acc1[i]);
      }
    }
  }
}

// ---------------------------------------------------------------------------
// Stage 2: TP1 fused with scatter. 4 waves/block, 32 edges/wave.
// Per-lane CG contractions -> LDS A-stage -> V_WMMA_F32_16X16X4_F32 fanout
// -> atomic scatter into nodeB.
// ---------------------------------------------------------------------------
__global__ __launch_bounds__(128) void k_tp1(
    const float* __restrict__ pos, const int* __restrict__ src, const int* __restrict__ dst,
    const float* __restrict__ nodeA, float* __restrict__ nodeB,
    const float* __restrict__ w3j, const float* __restrict__ w1m, int E) {
  __shared__ float sA[4][3200];
  __shared__ int sD[4][32];
  const int wave = threadIdx.x >> 5;
  const int lane = threadIdx.x & 31;
  float* A = sA[wave];
  int* D = sD[wave];
  long e = (long)blockIdx.x * 128 + wave * 32 + lane;
  bool valid = (e < E);
  int ec = valid ? (int)e : (E - 1);
  for (int i = lane; i < 3200; i += 32) A[i] = 0.f;

  int s_ = src[ec], d_ = dst[ec];
  D[lane] = valid ? d_ : -1;
  float sh[16];
  {
    float x = pos[3 * s_] - pos[3 * d_];
    float y = pos[3 * s_ + 1] - pos[3 * d_ + 1];
    float z = pos[3 * s_ + 2] - pos[3 * d_ + 2];
    calc_sh(x, y, z, sh);
  }
  float x1[16];
  for (int i = 0; i < 16; i++) x1[i] = valid ? nodeA[(size_t)s_ * 16 + i] : 0.f;

  // 24 CG paths: l1, l2, lo, output irrep c, path index within c
  static const int PL1[24] = {0,0,0,1,1,1,1,1,1,1,2,2,2,2,2,2,2,2,3,3,3,3,3,3};
  static const int PL2[24] = {0,1,2,0,1,1,1,2,2,3,0,1,1,2,2,2,3,3,1,2,2,3,3,3};
  static const int PLO[24] = {0,1,2,1,0,1,2,1,2,2,2,1,2,0,1,2,1,2,2,1,2,0,1,2};
  static const int PC [24] = {0,2,3,2,0,1,3,2,4,3,3,2,4,0,1,3,2,4,3,2,4,0,1,3};
  static const int PIX[24] = {0,0,0,1,1,0,1,2,0,2,3,3,1,2,1,4,4,2,5,5,3,3,2,6};
  static const int AOFFt[5] = {0, 128, 512, 1280, 2560};
  static const int PPADt[5] = {4, 4, 8, 8, 4};
  static const int LOFF[4] = {0, 1, 4, 9};

  for (int p = 0; p < 24; p++) {
    int l1 = PL1[p], l2 = PL2[p], lo = PLO[p], c = PC[p];
    const float* C = w3j + (((l1 * 4 + l2) * 3 + lo) * 245);
    int n1 = 2 * l1 + 1, n2 = 2 * l2 + 1, nk = 2 * lo + 1;
    const float* xa = x1 + LOFF[l1];
    const float* xb = sh + LOFF[l2];
    for (int k = 0; k < nk; k++) {
      float acc = 0.f;
      for (int i = 0; i < n1; i++) {
        float xv = xa[i];
        for (int j = 0; j < n2; j++)
          acc += xv * xb[j] * C[(i * 7 + j) * 5 + k];
      }
      A[AOFFt[c] + (k * 32 + lane) * PPADt[c] + PIX[p]] = acc;
    }
  }
  __syncthreads();

  const int h = lane >> 4, m = lane & 15;
  //                NK NT PPAD MO  OO   AOFF  W1O
  tp1_fanout<1, 4, 4, 64, 0, 0, 0>(A, w1m, D, nodeB, h, m);          // c0: 64x l=0
  tp1_fanout<3, 2, 4, 24, 64, 128, 256>(A, w1m, D, nodeB, h, m);     // c1: 24x l=1 even
  tp1_fanout<3, 2, 8, 24, 136, 512, 384>(A, w1m, D, nodeB, h, m);    // c2: 24x l=1 odd
  tp1_fanout<5, 1, 8, 16, 208, 1280, 640>(A, w1m, D, nodeB, h, m);   // c3: 16x l=2 even
  tp1_fanout<5, 1, 4, 16, 288, 2560, 768>(A, w1m, D, nodeB, h, m);   // c4: 16x l=2 odd
}

// ---------------------------------------------------------------------------
// Stage 3: TP2 fused with scatter. 2 waves/block, 32 edges/wave.
// 144 per-edge invariants -> LDS -> 36 chained K=4 WMMAs (dual M-half chains
// sharing each B fragment) -> nodeC scatter.
// ---------------------------------------------------------------------------
__global__ __launch_bounds__(64) void k_tp2(
    const float* __restrict__ pos, const int* __restrict__ src, const int* __restrict__ dst,
    const float* __restrict__ nodeB, float* __restrict__ nodeC,
    const float* __restrict__ w3j, const float* __restrict__ w2m, int E) {
  __shared__ float sA[2][32 * 145];
  __shared__ int sD[2][32];
  const int wave = threadIdx.x >> 5;
  const int lane = threadIdx.x & 31;
  float* A = sA[wave];
  int* D = sD[wave];
  long e = (long)blockIdx.x * 64 + wave * 32 + lane;
  bool valid = (e < E);
  int ec = valid ? (int)e : (E - 1);
  int s_ = src[ec], d_ = dst[ec];
  D[lane] = valid ? d_ : -1;
  float sh[16];
  {
    float x = pos[3 * s_] - pos[3 * d_];
    float y = pos[3 * s_ + 1] - pos[3 * d_ + 1];
    float z = pos[3 * s_ + 2] - pos[3 * d_ + 2];
    calc_sh(x, y, z, sh);
  }
  const float* nb = nodeB + (size_t)s_ * 368;
  const float C000 = w3j[0];                       // slot (0,0,0), k=0
  const float* C110 = w3j + ((1 * 4 + 1) * 3 + 0) * 245;
  const float* C220 = w3j + ((2 * 4 + 2) * 3 + 0) * 245;
  float* Ar = A + lane * 145;
  float sc = valid ? 1.f : 0.f;

  for (int u = 0; u < 24; u++) {        // q2: (24,1,+) x sh_l1 -> l0
    float acc = 0.f;
    for (int i = 0; i < 3; i++)
      for (int j = 0; j < 3; j++)
        acc += nb[64 + u * 3 + i] * sh[1 + j] * C110[(i * 7 + j) * 5];
    Ar[u] = acc * sc;
  }
  for (int u = 0; u < 16; u++) {        // q5: (16,2,-) x sh_l2 -> l0
    float acc = 0.f;
    for (int i = 0; i < 5; i++)
      for (int j = 0; j < 5; j++)
        acc += nb[288 + u * 5 + i] * sh[4 + j] * C220[(i * 7 + j) * 5];
    Ar[24 + u] = acc * sc;
  }
  for (int u = 0; u < 64; u++)          // q1: (64,0,+) x sh_l0 -> l0
    Ar[40 + u] = nb[u] * sh[0] * C000 * sc;
  for (int u = 0; u < 24; u++) {        // q3: (24,1,-) x sh_l1 -> l0
    float acc = 0.f;
    for (int i = 0; i < 3; i++)
      for (int j = 0; j < 3; j++)
        acc += nb[136 + u * 3 + i] * sh[1 + j] * C110[(i * 7 + j) * 5];
    Ar[104 + u] = acc * sc;
  }
  for (int u = 0; u < 16; u++) {        // q4: (16,2,+) x sh_l2 -> l0
    float acc = 0.f;
    for (int i = 0; i < 5; i++)
      for (int j = 0; j < 5; j++)
        acc += nb[208 + u * 5 + i] * sh[4 + j] * C220[(i * 7 + j) * 5];
    Ar[128 + u] = acc * sc;
  }
  Ar[144] = 0.f;
  __syncthreads();

  const int h = lane >> 4, m = lane & 15;
  v8f acc0 = {}, acc1 = {};
  for (int ks = 0; ks < 36; ks++) {
    int K0 = ks * 4 + 2 * h;
    v2f b, a0, a1;
    b[0] = w2m[K0 * 16 + m];
    b[1] = w2m[(K0 + 1) * 16 + m];
    const float* Ap0 = A + m * 145;
    const float* Ap1 = A + (16 + m) * 145;
    a0[0] = Ap0[K0]; a0[1] = Ap0[K0 + 1];
    a1[0] = Ap1[K0]; a1[1] = Ap1[K0 + 1];
    acc0 = __builtin_amdgcn_wmma_f32_16x16x4_f32(false, a0, false, b, (short)0, acc0,
                                                 false, false);
    acc1 = __builtin_amdgcn_wmma_f32_16x16x4_f32(false, a1, false, b, (short)0, acc1,
                                                 false, false);
  }
#pragma unroll
  for (int i = 0; i < 8; i++) {
    int r = i + 8 * h;
    int d0 = D[r];
    int d1 = D[16 + r];
    if (m < 7) {
      if (d0 >= 0) atomicAdd(nodeC + (size_t)d0 * 7 + m, acc0[i]);
      if (d1 >= 0) atomicAdd(nodeC + (size_t)d1 * 7 + m, acc1[i]);
    }
  }
}

// ---------------------------------------------------------------------------
// Stage 4: per-graph reduction (all scalings folded into w2m already)
// ---------------------------------------------------------------------------
__global__ void k_batch(const int* __restrict__ batch, const float* __restrict__ nodeC,
                        float* __restrict__ out, int N) {
  int n = blockIdx.x * blockDim.x + threadIdx.x;
  if (n >= N) return;
  int g = batch[n];
  for (int j = 0; j < 7; j++)
    atomicAdd(out + (size_t)g * 7 + j, nodeC[(size_t)n * 7 + j]);
}

// ---------------------------------------------------------------------------
extern "C" void kernel_launch(void* const* d_in, const int* in_sizes, int n_in,
                              void* d_out, int out_size, void* d_ws, size_t ws_size,
                              hipStream_t stream) {
  const float* pos = (const float*)d_in[0];
  const float* w1 = (const float*)d_in[1];
  const float* w2 = (const float*)d_in[2];
  const int* esrc = (const int*)d_in[3];
  const int* edst = (const int*)d_in[4];
  const int* batch = (const int*)d_in[5];
  int nNodes = in_sizes[0] / 3;
  int E = in_sizes[3];

  float* wsf = (float*)d_ws;
  float* w3j = wsf;                              // 48*245 = 11760
  float* w1m = w3j + 48 * 245;                   // 832
  float* w2m = w1m + 832;                        // 2304
  float* nodeA = w2m + 2304;                     // nNodes*16
  float* nodeB = nodeA + (size_t)nNodes * 16;    // nNodes*368
  float* nodeC = nodeB + (size_t)nNodes * 368;   // nNodes*7

  hipMemsetAsync(nodeA, 0, (size_t)nNodes * 16 * sizeof(float), stream);
  hipMemsetAsync(nodeB, 0, (size_t)nNodes * 368 * sizeof(float), stream);
  hipMemsetAsync(nodeC, 0, (size_t)nNodes * 7 * sizeof(float), stream);
  hipMemsetAsync(d_out, 0, (size_t)out_size * sizeof(float), stream);

  k_build_w3j<<<1, 64, 0, stream>>>(w3j);
  k_prep_weights<<<(832 + 2304 + 255) / 256, 256, 0, stream>>>(w1, w2, w1m, w2m);
  k_sh_scatter<<<(E + 255) / 256, 256, 0, stream>>>(pos, esrc, edst, nodeA, E);
  k_tp1<<<(E + 127) / 128, 128, 0, stream>>>(pos, esrc, edst, nodeA, nodeB, w3j, w1m, E);
  k_tp2<<<(E + 63) / 64, 64, 0, stream>>>(pos, esrc, edst, nodeB, nodeC, w3j, w2m, E);
  k_batch<<<(nNodes + 255) / 256, 256, 0, stream>>>(batch, nodeC, (float*)d_out, nNodes);
}